// Codebook_83348135346489
// MI455X (gfx1250) — compile-verified
//
#include <hip/hip_runtime.h>

typedef __attribute__((ext_vector_type(2))) float v2f;
typedef __attribute__((ext_vector_type(4))) float v4f;
typedef __attribute__((ext_vector_type(8))) float v8f;

#define NUM_CODES   1024
#define EMBED_DIM   256
#define HW          1024            // 32*32 spatial per batch image
#define NROWS       32768           // 32 * 32 * 32
#define ROWS_PER_WG 128
#define NTILES      (NUM_CODES / 16)  // 64 code tiles
#define KSTEPS      (EMBED_DIM / 4)   // 64 wmma k-steps
#define EPAD        260               // padded LDS row stride (dwords): bank-conflict-free b64 reads

#define ZQ_OFF  0
#define CB_OFF  8388608
#define CM_OFF  8388609
#define IDX_OFF 8388610
#define N_ELEM  8388608.0f

__global__ __launch_bounds__(256, 1)
void vq_dist_argmin_gather(const float* __restrict__ z,
                           const float* __restrict__ emb,
                           float* __restrict__ out,
                           float* __restrict__ ws) {
  __shared__ float ebuf[2][16 * EPAD];
  __shared__ float enorm[2][16];
  __shared__ int   idx_lds[ROWS_PER_WG];
  __shared__ float wsum[8];

  const int t    = threadIdx.x;
  const int lane = t & 31;
  const int w    = t >> 5;        // wave id (8 waves)
  const int h    = lane >> 4;     // half-wave
  const int m    = lane & 15;

  const int R0   = (int)blockIdx.x * ROWS_PER_WG;  // block row base
  const int r0   = R0 + w * 16;                    // wave row base (16 rows per wave)
  const int b    = r0 >> 10;                       // batch index (128 | 1024)
  const int hw0  = r0 & 1023;
  const int zbase = b * EMBED_DIM * HW + hw0;      // + c*HW + m

  // ---- Phase A: A fragments: z rows r0..r0+15, full K=256, coalesced, kept in VGPRs.
  // 16x4 fp32 A layout: lanes 0-15 hold {K=0,K=1}, lanes 16-31 hold {K=2,K=3}.
  v2f a[KSTEPS];
#pragma unroll
  for (int k = 0; k < KSTEPS; ++k) {
    const int c = 4 * k + 2 * h;
    a[k].x = z[zbase + c * HW + m];
    a[k].y = z[zbase + (c + 1) * HW + m];
  }

  float minS[8];
  int   minI[8];
#pragma unroll
  for (int j = 0; j < 8; ++j) { minS[j] = 3.0e38f; minI[j] = 0; }

  const int ss_row = t >> 4;   // staging: code row within tile (16 threads per row)
  const int ss_col = t & 15;

  // Stage split: global loads issued early (stage_ld), LDS writes + norms after compute.
  auto stage_ld = [&](int nt, v4f* st) {
    const float* src = emb + (nt * 16 + ss_row) * EMBED_DIM + ss_col * 16;
#pragma unroll
    for (int u = 0; u < 4; ++u) st[u] = *(const v4f*)(src + 4 * u);
  };
  auto stage_st = [&](const v4f* st, int buf) {
    float* dst = &ebuf[buf][ss_row * EPAD + ss_col * 16];
    float ss = 0.0f;
#pragma unroll
    for (int u = 0; u < 4; ++u) {
      *(v4f*)(dst + 4 * u) = st[u];
      ss += st[u].x * st[u].x + st[u].y * st[u].y + st[u].z * st[u].z + st[u].w * st[u].w;
    }
#pragma unroll
    for (int msk = 8; msk >= 1; msk >>= 1) ss += __shfl_xor(ss, msk, 16);
    if (ss_col == 0) enorm[buf][ss_row] = ss;
  };

  v4f st[4];
  stage_ld(0, st);
  stage_st(st, 0);
  __syncthreads();

  // ---- Phase B: 64 code tiles; LDS double-buffered across tiles, B fragments
  // double-buffered in registers inside the k-loop, two accumulators.
  for (int nt = 0; nt < NTILES; ++nt) {
    const int buf = nt & 1;
    const bool more = (nt + 1 < NTILES);
    if (more) stage_ld(nt + 1, st);          // global loads in flight during WMMAs

    v8f acc0 = {0.f, 0.f, 0.f, 0.f, 0.f, 0.f, 0.f, 0.f};
    v8f acc1 = {0.f, 0.f, 0.f, 0.f, 0.f, 0.f, 0.f, 0.f};
    // B fragment (4x16 fp32): lane holds e[n0+m][4k+2h .. +1] -> b64 from padded LDS.
    const float* eb = &ebuf[buf][m * EPAD + 2 * h];

    v2f bfA[8], bfB[8];
#pragma unroll
    for (int k = 0; k < 8; ++k) bfA[k] = *(const v2f*)(eb + 4 * k);

#pragma unroll
    for (int kb = 0; kb < KSTEPS; kb += 16) {
#pragma unroll
      for (int k = 0; k < 8; ++k) bfB[k] = *(const v2f*)(eb + 4 * (kb + 8 + k));
#pragma unroll
      for (int k = 0; k < 8; ++k) {
        if (k & 1) acc1 = __builtin_amdgcn_wmma_f32_16x16x4_f32(
            false, a[kb + k], false, bfA[k], (short)0, acc1, false, false);
        else       acc0 = __builtin_amdgcn_wmma_f32_16x16x4_f32(
            false, a[kb + k], false, bfA[k], (short)0, acc0, false, false);
      }
      if (kb + 16 < KSTEPS) {
#pragma unroll
        for (int k = 0; k < 8; ++k) bfA[k] = *(const v2f*)(eb + 4 * (kb + 16 + k));
      }
#pragma unroll
      for (int k = 0; k < 8; ++k) {
        if (k & 1) acc1 = __builtin_amdgcn_wmma_f32_16x16x4_f32(
            false, a[kb + 8 + k], false, bfB[k], (short)0, acc1, false, false);
        else       acc0 = __builtin_amdgcn_wmma_f32_16x16x4_f32(
            false, a[kb + 8 + k], false, bfB[k], (short)0, acc0, false, false);
      }
    }
    const v8f acc = acc0 + acc1;

    const float nrm  = enorm[buf][m];
    const int n_idx  = nt * 16 + m;
#pragma unroll
    for (int j = 0; j < 8; ++j) {
      float s = nrm - 2.0f * acc[j];       // ||e||^2 - 2 z.e  (||z||^2 is row-constant)
      if (s < minS[j]) { minS[j] = s; minI[j] = n_idx; }  // strict '<' keeps first index
    }

    if (more) stage_st(st, buf ^ 1);       // waits loads, fills other LDS buffer
    __syncthreads();
  }

  // ---- Phase C: argmin across the 16 lanes holding each row's candidates.
#pragma unroll
  for (int msk = 1; msk < 16; msk <<= 1) {
#pragma unroll
    for (int j = 0; j < 8; ++j) {
      float oS = __shfl_xor(minS[j], msk, 16);
      int   oI = __shfl_xor(minI[j], msk, 16);
      if (oS < minS[j] || (oS == minS[j] && oI < minI[j])) { minS[j] = oS; minI[j] = oI; }
    }
  }
  if (m == 0) {
#pragma unroll
    for (int j = 0; j < 8; ++j) {
      const int row = j + 8 * h;           // C/D layout: VGPR j, half h -> M = j + 8h
      idx_lds[w * 16 + row] = minI[j];
      out[IDX_OFF + r0 + row] = (float)minI[j];
    }
  }
  __syncthreads();

  // ---- Phase D: gather z_q, straight-through output, loss partial. Coalesced writes.
  const int rr   = t & 127;                 // row offset within block tile
  const int ch0  = t >> 7;                  // 0/1: even/odd channels
  const int code = idx_lds[rr];
  const float* erow = emb + code * EMBED_DIM;
  const int zb2 = (R0 >> 10) * EMBED_DIM * HW + (R0 & 1023) + rr;

  float lsum = 0.0f;
  for (int c = ch0; c < EMBED_DIM; c += 2) {
    const int off = zb2 + c * HW;
    const float zv = z[off];
    const float q  = erow[c];
    __builtin_nontemporal_store(zv + (q - zv), &out[ZQ_OFF + off]);  // straight-through
    const float d = q - zv;
    lsum += d * d;
  }

#pragma unroll
  for (int msk = 16; msk >= 1; msk >>= 1) lsum += __shfl_xor(lsum, msk, 32);
  if (lane == 0) wsum[w] = lsum;
  __syncthreads();
  if (t == 0) {
    float s = 0.0f;
#pragma unroll
    for (int i = 0; i < 8; ++i) s += wsum[i];
    ws[blockIdx.x] = s;                     // plain store: deterministic across replays
  }
}

__global__ void vq_loss_reduce(const float* __restrict__ ws, float* __restrict__ out) {
  __shared__ float sm[8];
  const int t = threadIdx.x;
  float v = ws[t];
#pragma unroll
  for (int msk = 16; msk >= 1; msk >>= 1) v += __shfl_xor(v, msk, 32);
  if ((t & 31) == 0) sm[t >> 5] = v;
  __syncthreads();
  if (t == 0) {
    float s = 0.0f;
#pragma unroll
    for (int i = 0; i < 8; ++i) s += sm[i];
    const float mean = s / N_ELEM;
    out[CB_OFF] = mean;                     // codebook loss
    out[CM_OFF] = 0.25f * mean;             // commitment loss (same forward value * beta)
  }
}

extern "C" void kernel_launch(void* const* d_in, const int* in_sizes, int n_in,
                              void* d_out, int out_size, void* d_ws, size_t ws_size,
                              hipStream_t stream) {
  const float* z   = (const float*)d_in[0];
  const float* emb = (const float*)d_in[1];
  float* out = (float*)d_out;
  float* ws  = (float*)d_ws;
  (void)in_sizes; (void)n_in; (void)out_size; (void)ws_size;

  vq_dist_argmin_gather<<<NROWS / ROWS_PER_WG, 256, 0, stream>>>(z, emb, out, ws);
  vq_loss_reduce<<<1, 256, 0, stream>>>(ws, out);
}